// EncNP_57174604644729
// MI455X (gfx1250) — compile-verified
//
#include <hip/hip_runtime.h>
#include <math.h>

typedef __attribute__((ext_vector_type(2))) float v2f;
typedef __attribute__((ext_vector_type(8))) float v8f;

#define NBATCH 4
#define NPTS   4096
#define EMBED  72
#define K_NB   90
#define ALPHA_C 1000.0f
#define BETA_C  100.0f
#define EPS_C   1e-5f
#define LOG_ALPHA 6.90775527898f   // ln(1000)

// ---------------------------------------------------------------------------
// Initial positional embedding: x [B,3,N] -> feat [B,N,72] (point-major)
// channel c = ci*24 + f*2 + parity (parity 0 = sin, 1 = cos), fd = 12
// ---------------------------------------------------------------------------
__global__ void k_pose_init(const float* __restrict__ x, float* __restrict__ feat) {
    int i = blockIdx.x * blockDim.x + threadIdx.x;
    const int total = NBATCH * NPTS * EMBED;
    if (i >= total) return;
    int c = i % EMBED;
    int n = (i / EMBED) % NPTS;
    int b = i / (EMBED * NPTS);
    int ci  = c / 24;
    int rem = c % 24;
    int f   = rem >> 1;
    int par = rem & 1;
    float coord = x[(b * 3 + ci) * NPTS + n];
    float scale = BETA_C * __expf(-LOG_ALPHA * ((float)f / 12.0f));
    float dv = coord * scale;
    feat[(b * NPTS + n) * EMBED + c] = par ? __cosf(dv) : __sinf(dv);
}

// ---------------------------------------------------------------------------
// Farthest point sampling on batch 0: xyz [N,3] -> idx[S]. One block.
// Points + running min-dist live in registers; argmax via LDS tree reduce.
// ---------------------------------------------------------------------------
#define FPS_T 512
__global__ void __launch_bounds__(FPS_T) k_fps(const float* __restrict__ xyz,
                                               int* __restrict__ idx, int N, int S) {
    __shared__ float rv[FPS_T];
    __shared__ int   ri[FPS_T];
    __shared__ float selxyz[3];
    int tid = threadIdx.x;
    const int P = 8;  // 4096 / 512
    float px[P], py[P], pz[P], dd[P];
    int np = 0;
    for (int p = 0; p < P; ++p) {
        int n = tid + p * FPS_T;
        if (n < N) {
            px[p] = xyz[n * 3 + 0];
            py[p] = xyz[n * 3 + 1];
            pz[p] = xyz[n * 3 + 2];
            dd[p] = 3.4e38f;
            np = p + 1;
        }
    }
    int sel = 0;
    if (tid == 0) { selxyz[0] = xyz[0]; selxyz[1] = xyz[1]; selxyz[2] = xyz[2]; }
    __syncthreads();
    for (int t = 0; t < S; ++t) {
        if (tid == 0) idx[t] = sel;
        float sx = selxyz[0], sy = selxyz[1], sz = selxyz[2];
        float bv = -1.0f; int bi = 0x7fffffff;
        for (int p = 0; p < np; ++p) {
            float ex = px[p] - sx, ey = py[p] - sy, ez = pz[p] - sz;
            float dist = ex * ex + ey * ey + ez * ez;
            dd[p] = fminf(dd[p], dist);
            int n = tid + p * FPS_T;
            if (dd[p] > bv || (dd[p] == bv && n < bi)) { bv = dd[p]; bi = n; }
        }
        rv[tid] = bv; ri[tid] = bi;
        __syncthreads();
        for (int st = FPS_T / 2; st > 0; st >>= 1) {
            if (tid < st) {
                if (rv[tid + st] > rv[tid] ||
                    (rv[tid + st] == rv[tid] && ri[tid + st] < ri[tid])) {
                    rv[tid] = rv[tid + st]; ri[tid] = ri[tid + st];
                }
            }
            __syncthreads();
        }
        sel = ri[0];
        if (tid == 0) {
            selxyz[0] = xyz[sel * 3 + 0];
            selxyz[1] = xyz[sel * 3 + 1];
            selxyz[2] = xyz[sel * 3 + 2];
        }
        __syncthreads();
    }
}

// ---------------------------------------------------------------------------
// Gather sampled centers: lc_x [B,S,C], xyzNext [B,S,3]
// ---------------------------------------------------------------------------
__global__ void k_gather(const float* __restrict__ featPrev, const float* __restrict__ xyzPrev,
                         const int* __restrict__ fpsIdx, float* __restrict__ lcx,
                         float* __restrict__ xyzNext, int S, int C, int Nprev) {
    int i = blockIdx.x * blockDim.x + threadIdx.x;
    if (i < NBATCH * S * C) {
        int c = i % C; int s = (i / C) % S; int b = i / (C * S);
        int n = fpsIdx[s];
        lcx[(b * S + s) * C + c] = featPrev[(b * Nprev + n) * C + c];
    }
    if (i < NBATCH * S * 3) {
        int c3 = i % 3; int s = (i / 3) % S; int b = i / (3 * S);
        int n = fpsIdx[s];
        xyzNext[(b * S + s) * 3 + c3] = xyzPrev[(b * Nprev + n) * 3 + c3];
    }
}

// ---------------------------------------------------------------------------
// kNN via WMMA F32 16x16x4: one wave handles 16 queries, streams the DB in
// 16-point chunks; Gram matrix on the matrix pipe; -2*dot + |q|^2 + |p|^2.
// Top-90 per query as unsorted list with tracked worst (order irrelevant
// downstream: max/mean over k and global std are permutation-invariant).
// ---------------------------------------------------------------------------
__global__ void __launch_bounds__(32) k_knn(const float* __restrict__ xyzDB,
                                            const float* __restrict__ xyzQ,
                                            int* __restrict__ knnIdx, int N, int S) {
    int lid = threadIdx.x;
    int nt  = S / 16;
    int qt  = blockIdx.x % nt;
    int b   = blockIdx.x / nt;
    int q0  = qt * 16;
    __shared__ float qn2[16];
    __shared__ float tile[256];
    __shared__ float topd[16 * K_NB];
    __shared__ int   topi[16 * K_NB];

    int lm = lid & 15;
    const float* qp = xyzQ + (size_t)(b * S + q0 + lm) * 3;
    float qx = qp[0], qy = qp[1], qz = qp[2];
    if (lid < 16) qn2[lid] = qx * qx + qy * qy + qz * qz;

    float worstV = 3.4e38f; int worstP = 0;
    if (lid < 16) {
        for (int j = 0; j < K_NB; ++j) { topd[lid * K_NB + j] = 3.4e38f; topi[lid * K_NB + j] = 0; }
    }
    // A operand (16x4, K=3 padded with 0): lanes 0-15 hold K=0,1; lanes 16-31 hold K=2,3
    v2f amat; amat.x = (lid < 16) ? qx : qz; amat.y = (lid < 16) ? qy : 0.0f;
    __syncthreads();

    for (int c0 = 0; c0 < N; c0 += 16) {
        const float* pp = xyzDB + (size_t)(b * N + c0 + lm) * 3;
        if (c0 + 16 < N) __builtin_prefetch(xyzDB + (size_t)(b * N + c0 + 16 + lm) * 3, 0, 1);
        float px = pp[0], py = pp[1], pz = pp[2];
        float pn2 = px * px + py * py + pz * pz;
        // B operand (4x16): lanes 0-15 hold K=0,1 rows; lanes 16-31 hold K=2,3
        v2f bmat; bmat.x = (lid < 16) ? px : pz; bmat.y = (lid < 16) ? py : 0.0f;
        v8f acc = {0.f, 0.f, 0.f, 0.f, 0.f, 0.f, 0.f, 0.f};
        acc = __builtin_amdgcn_wmma_f32_16x16x4_f32(false, amat, false, bmat,
                                                    (short)0, acc, false, false);
        int mhi = (lid >= 16) ? 8 : 0;
#pragma unroll
        for (int r = 0; r < 8; ++r) {
            int m = r + mhi;
            float dist = qn2[m] + pn2 - 2.0f * acc[r];
            tile[m * 16 + lm] = dist;
        }
        __syncthreads();
        if (lid < 16) {
            float* myd = topd + lid * K_NB;
            int*   myi = topi + lid * K_NB;
            for (int j = 0; j < 16; ++j) {
                float dv = tile[lid * 16 + j];
                if (dv < worstV) {
                    myd[worstP] = dv; myi[worstP] = c0 + j;
                    float wv = -1.0f; int wp = 0;
                    for (int t = 0; t < K_NB; ++t) {
                        float v = myd[t];
                        if (v > wv) { wv = v; wp = t; }
                    }
                    worstV = wv; worstP = wp;
                }
            }
        }
        __syncthreads();
    }
    if (lid < 16) {
        for (int j = 0; j < K_NB; ++j)
            knnIdx[(size_t)(b * S + q0 + lid) * K_NB + j] = topi[lid * K_NB + j];
    }
}

// ---------------------------------------------------------------------------
// Global std (ddof=1) of dx and dxyz: deterministic 2-pass reduction.
// ---------------------------------------------------------------------------
__global__ void __launch_bounds__(128) k_stats_partial(
    const float* __restrict__ featPrev, const float* __restrict__ lcx,
    const float* __restrict__ xyzPrev, const float* __restrict__ xyzNext,
    const int* __restrict__ knnIdx, float* __restrict__ part,
    int S, int C, int Nprev) {
    int i = blockIdx.x * blockDim.x + threadIdx.x;
    int total = NBATCH * S * K_NB;
    float s1 = 0.f, s2 = 0.f, t1 = 0.f, t2 = 0.f;
    if (i < total) {
        int s = (i / K_NB) % S; int b = i / (K_NB * S);
        int n = knnIdx[i];
        const float* fp = featPrev + (size_t)(b * Nprev + n) * C;
        const float* lp = lcx + (size_t)(b * S + s) * C;
        for (int c = 0; c < C; ++c) { float d = fp[c] - lp[c]; s1 += d; s2 += d * d; }
        const float* xp = xyzPrev + (size_t)(b * Nprev + n) * 3;
        const float* xq = xyzNext + (size_t)(b * S + s) * 3;
        for (int c = 0; c < 3; ++c) { float d = xp[c] - xq[c]; t1 += d; t2 += d * d; }
    }
    __shared__ float red[4][128];
    int tid = threadIdx.x;
    red[0][tid] = s1; red[1][tid] = s2; red[2][tid] = t1; red[3][tid] = t2;
    __syncthreads();
    for (int st = 64; st > 0; st >>= 1) {
        if (tid < st)
            for (int u = 0; u < 4; ++u) red[u][tid] += red[u][tid + st];
        __syncthreads();
    }
    if (tid == 0) {
        int bo = blockIdx.x * 4;
        part[bo + 0] = red[0][0]; part[bo + 1] = red[1][0];
        part[bo + 2] = red[2][0]; part[bo + 3] = red[3][0];
    }
}

__global__ void __launch_bounds__(256) k_stats_finish(const float* __restrict__ part,
                                                      int nblocks, float* __restrict__ scal,
                                                      float nx, float nxyz) {
    __shared__ float red[4][256];
    int tid = threadIdx.x;
    float a0 = 0.f, a1 = 0.f, a2 = 0.f, a3 = 0.f;
    for (int i = tid; i < nblocks; i += 256) {
        a0 += part[i * 4 + 0]; a1 += part[i * 4 + 1];
        a2 += part[i * 4 + 2]; a3 += part[i * 4 + 3];
    }
    red[0][tid] = a0; red[1][tid] = a1; red[2][tid] = a2; red[3][tid] = a3;
    __syncthreads();
    for (int st = 128; st > 0; st >>= 1) {
        if (tid < st)
            for (int u = 0; u < 4; ++u) red[u][tid] += red[u][tid + st];
        __syncthreads();
    }
    if (tid == 0) {
        float sum = red[0][0], sq = red[1][0];
        float var = (sq - sum * sum / nx) / (nx - 1.0f);
        scal[0] = 1.0f / (sqrtf(fmaxf(var, 0.f)) + EPS_C);
        sum = red[2][0]; sq = red[3][0];
        var = (sq - sum * sum / nxyz) / (nxyz - 1.0f);
        scal[1] = 1.0f / (sqrtf(fmaxf(var, 0.f)) + EPS_C);
    }
}

// ---------------------------------------------------------------------------
// Fused PE + feature + (max,mean over k) -> lcbuf [D, B*S] (channel-major).
// Block per (b,s); channels held in registers; w never materialized.
// ---------------------------------------------------------------------------
__global__ void __launch_bounds__(128) k_stage_main(
    const float* __restrict__ featPrev, const float* __restrict__ lcx,
    const float* __restrict__ xyzPrev, const float* __restrict__ xyzNext,
    const int* __restrict__ knnIdx, const float* __restrict__ scal,
    float* __restrict__ lcbuf, int S, int C, int Nprev, int D, int fd) {
    int bs = blockIdx.x;
    int s = bs % S; int b = bs / S;
    int tid = threadIdx.x;
    const int MAXR = 9;
    int R = (D + 127) / 128;
    float scale_r[MAXR]; int cidx_r[MAXR]; int par_r[MAXR];
    float wmax[MAXR], wsum[MAXR];
    int twofd = 2 * fd;
    for (int r = 0; r < MAXR; ++r) { wmax[r] = -3.4e38f; wsum[r] = 0.f; }
    for (int r = 0; r < R; ++r) {
        int d = tid + r * 128;
        if (d < D) {
            cidx_r[r] = d / twofd;
            int rem = d % twofd;
            int f = rem >> 1; par_r[r] = rem & 1;
            scale_r[r] = BETA_C * __expf(-LOG_ALPHA * ((float)f / (float)fd));
        }
    }
    __shared__ float lcx_sh[576];
    __shared__ float dn[3];
    __shared__ int   nidx_sh;
    for (int c = tid; c < C; c += 128) lcx_sh[c] = lcx[(size_t)(b * S + s) * C + c];
    float invx = scal[0], invxyz = scal[1];
    float qx = xyzNext[(size_t)(b * S + s) * 3 + 0];
    float qy = xyzNext[(size_t)(b * S + s) * 3 + 1];
    float qz = xyzNext[(size_t)(b * S + s) * 3 + 2];
    __syncthreads();
    for (int k = 0; k < K_NB; ++k) {
        if (tid == 0) {
            int n = knnIdx[(size_t)(b * S + s) * K_NB + k];
            nidx_sh = n;
            dn[0] = (xyzPrev[(size_t)(b * Nprev + n) * 3 + 0] - qx) * invxyz;
            dn[1] = (xyzPrev[(size_t)(b * Nprev + n) * 3 + 1] - qy) * invxyz;
            dn[2] = (xyzPrev[(size_t)(b * Nprev + n) * 3 + 2] - qz) * invxyz;
        }
        __syncthreads();
        int n = nidx_sh;
        const float* fp = featPrev + (size_t)(b * Nprev + n) * C;
        for (int r = 0; r < R; ++r) {
            int d = tid + r * 128;
            if (d < D) {
                float dv = dn[cidx_r[r]] * scale_r[r];
                float pe = par_r[r] ? __cosf(dv) : __sinf(dv);
                float fv = (d < C) ? (fp[d] - lcx_sh[d]) * invx : lcx_sh[d - C];
                float w = (fv + pe) * pe;
                wmax[r] = fmaxf(wmax[r], w);
                wsum[r] += w;
            }
        }
        __syncthreads();
    }
    int M = NBATCH * S;
    for (int r = 0; r < R; ++r) {
        int d = tid + r * 128;
        if (d < D)
            lcbuf[(size_t)d * M + b * S + s] = wmax[r] + wsum[r] * (1.0f / (float)K_NB);
    }
}

// ---------------------------------------------------------------------------
// BatchNorm over (B,S) per channel (ddof=0), then normalize+GELU+transpose.
// ---------------------------------------------------------------------------
__global__ void __launch_bounds__(256) k_bn_stats(const float* __restrict__ lcbuf,
                                                  float* __restrict__ mu,
                                                  float* __restrict__ rstd, int M) {
    int d = blockIdx.x; int tid = threadIdx.x;
    __shared__ float r1[256], r2[256];
    float a = 0.f, a2 = 0.f;
    for (int i = tid; i < M; i += 256) { float v = lcbuf[(size_t)d * M + i]; a += v; a2 += v * v; }
    r1[tid] = a; r2[tid] = a2;
    __syncthreads();
    for (int st = 128; st > 0; st >>= 1) {
        if (tid < st) { r1[tid] += r1[tid + st]; r2[tid] += r2[tid + st]; }
        __syncthreads();
    }
    if (tid == 0) {
        float m = r1[0] / (float)M;
        float var = r2[0] / (float)M - m * m;
        mu[d] = m;
        rstd[d] = rsqrtf(fmaxf(var, 0.f) + EPS_C);
    }
}

__global__ void k_bn_apply(const float* __restrict__ lcbuf, const float* __restrict__ mu,
                           const float* __restrict__ rstd, const float* __restrict__ gamma,
                           const float* __restrict__ beta, float* __restrict__ featNext,
                           int S, int D) {
    int i = blockIdx.x * blockDim.x + threadIdx.x;
    int M = NBATCH * S;
    if (i >= D * M) return;
    int idx = i % M; int d = i / M;
    float x = (lcbuf[(size_t)d * M + idx] - mu[d]) * rstd[d] * gamma[d] + beta[d];
    float g = 0.5f * x * (1.0f + erff(x * 0.70710678118f));
    featNext[(size_t)idx * D + d] = g;
}

// ---------------------------------------------------------------------------
// Final: max_s + mean_s over feat [B, 256, 1152] -> out [B, 1152]
// ---------------------------------------------------------------------------
__global__ void __launch_bounds__(256) k_final(const float* __restrict__ feat,
                                               float* __restrict__ out) {
    const int S = 256, D = 1152;
    int bd = blockIdx.x;
    int d = bd % D; int b = bd / D;
    int tid = threadIdx.x;
    float v = feat[(size_t)(b * S + tid) * D + d];
    __shared__ float rm[256], rs[256];
    rm[tid] = v; rs[tid] = v;
    __syncthreads();
    for (int st = 128; st > 0; st >>= 1) {
        if (tid < st) { rm[tid] = fmaxf(rm[tid], rm[tid + st]); rs[tid] += rs[tid + st]; }
        __syncthreads();
    }
    if (tid == 0) out[b * D + d] = rm[0] + rs[0] / (float)S;
}

// ---------------------------------------------------------------------------
extern "C" void kernel_launch(void* const* d_in, const int* in_sizes, int n_in,
                              void* d_out, int out_size, void* d_ws, size_t ws_size,
                              hipStream_t stream) {
    const float* xyz_in = (const float*)d_in[0];  // [4,4096,3]
    const float* x_in   = (const float*)d_in[1];  // [4,3,4096]
    const float* gam[4] = {(const float*)d_in[2], (const float*)d_in[4],
                           (const float*)d_in[6], (const float*)d_in[8]};
    const float* bet[4] = {(const float*)d_in[3], (const float*)d_in[5],
                           (const float*)d_in[7], (const float*)d_in[9]};
    (void)in_sizes; (void)n_in; (void)out_size; (void)ws_size;

    char* ws = (char*)d_ws;
    size_t off = 0;
    auto alloc = [&](size_t bytes) -> void* {
        void* p = ws + off;
        off += (bytes + 255) & ~(size_t)255;
        return p;
    };
    float* featA  = (float*)alloc((size_t)1179648 * 4);
    float* featB  = (float*)alloc((size_t)1179648 * 4);
    float* xyzA   = (float*)alloc((size_t)NBATCH * 2048 * 3 * 4);
    float* xyzB   = (float*)alloc((size_t)NBATCH * 2048 * 3 * 4);
    int*   fpsIdx = (int*)alloc((size_t)2048 * 4);
    int*   knnIdx = (int*)alloc((size_t)NBATCH * 2048 * K_NB * 4);
    float* lcx    = (float*)alloc((size_t)NBATCH * 2048 * 72 * 4);
    float* lcbuf  = (float*)alloc((size_t)1179648 * 4);
    float* part   = (float*)alloc((size_t)5760 * 4 * 4);
    float* scal   = (float*)alloc(64);
    float* muB    = (float*)alloc((size_t)1152 * 4);
    float* rstdB  = (float*)alloc((size_t)1152 * 4);

    { int total = NBATCH * NPTS * EMBED;
      k_pose_init<<<(total + 255) / 256, 256, 0, stream>>>(x_in, featA); }

    const float* featPrev = featA;
    float* featNext = featB;
    const float* xyzPrev = xyz_in;
    float* xyzNext = xyzA;
    int Nprev = NPTS;
    for (int st = 0; st < 4; ++st) {
        int S = Nprev / 2, C = EMBED << st, D = 144 << st, fd = D / 6;
        k_fps<<<1, FPS_T, 0, stream>>>(xyzPrev, fpsIdx, Nprev, S);
        { int tot = NBATCH * S * C;
          k_gather<<<(tot + 255) / 256, 256, 0, stream>>>(featPrev, xyzPrev, fpsIdx,
                                                          lcx, xyzNext, S, C, Nprev); }
        k_knn<<<NBATCH * (S / 16), 32, 0, stream>>>(xyzPrev, xyzNext, knnIdx, Nprev, S);
        { int tot = NBATCH * S * K_NB;
          int nb = (tot + 127) / 128;
          k_stats_partial<<<nb, 128, 0, stream>>>(featPrev, lcx, xyzPrev, xyzNext,
                                                  knnIdx, part, S, C, Nprev);
          float nx = (float)tot * (float)C;
          float nxyz = (float)tot * 3.0f;
          k_stats_finish<<<1, 256, 0, stream>>>(part, nb, scal, nx, nxyz); }
        k_stage_main<<<NBATCH * S, 128, 0, stream>>>(featPrev, lcx, xyzPrev, xyzNext,
                                                     knnIdx, scal, lcbuf, S, C, Nprev, D, fd);
        int M = NBATCH * S;
        k_bn_stats<<<D, 256, 0, stream>>>(lcbuf, muB, rstdB, M);
        { int tot = D * M;
          k_bn_apply<<<(tot + 255) / 256, 256, 0, stream>>>(lcbuf, muB, rstdB,
                                                            gam[st], bet[st], featNext, S, D); }
        featPrev = featNext;
        featNext = (featNext == featB) ? featA : featB;
        xyzPrev = xyzNext;
        xyzNext = (xyzNext == xyzA) ? xyzB : xyzA;
        Nprev = S;
    }
    k_final<<<NBATCH * 1152, 256, 0, stream>>>(featPrev, (float*)d_out);
}